// SparseFeatureExtractor_979252543683
// MI455X (gfx1250) — compile-verified
//
#include <hip/hip_runtime.h>

// ---------------------------------------------------------------------------
// SparseFeatureExtractor for MI455X (gfx1250), wave32.
//  - f32 WMMA 16x16x4 for exact-precision sparse-conv GEMMs
//  - weights staged into LDS via Tensor Data Mover (tensor_load_to_lds,
//    TENSORcnt) with a cooperative-copy backstop
// ---------------------------------------------------------------------------

typedef float v2f __attribute__((ext_vector_type(2)));
typedef float v8f __attribute__((ext_vector_type(8)));
typedef unsigned int u32x4 __attribute__((ext_vector_type(4)));
typedef int i32x4 __attribute__((ext_vector_type(4)));
typedef int i32x8 __attribute__((ext_vector_type(8)));

#define GB 8
#define GH 512
#define GW 512
#define EPS 1e-5f
#define SLOPE 0.01f

// -------------------------------- grid build -------------------------------
__global__ __launch_bounds__(256) void grid_init(int* __restrict__ grid, int total) {
  for (int i = blockIdx.x * blockDim.x + threadIdx.x; i < total;
       i += gridDim.x * blockDim.x)
    grid[i] = -1;
}

__global__ __launch_bounds__(256) void grid_scatter(const int* __restrict__ indices,
                                                    int* __restrict__ grid, int N) {
  int i = blockIdx.x * blockDim.x + threadIdx.x;
  if (i < N) {
    int b = indices[i * 3 + 0];
    int y = indices[i * 3 + 1];
    int x = indices[i * 3 + 2];
    grid[(b * GH + y) * GW + x] = i;
  }
}

// --------------------- TDM weight staging into LDS -------------------------
// 1-D DMA of `total` f32 elements from global `src` to LDS offset `lds_off`.
// D# per CDNA5 ISA: group0 = {count=1 | lds_addr | global_addr | type=2},
// group1 = {data_size=4B, tensor_dim0=total, tile_dim0=total (<=65535)}.
__device__ __forceinline__ void tdm_load_f32(unsigned lds_off,
                                             const float* __restrict__ src,
                                             unsigned total_elems) {
#if __has_builtin(__builtin_amdgcn_tensor_load_to_lds)
  unsigned long long ga = (unsigned long long)(const void*)src;
  u32x4 g0;
  g0.x = 1u;                                              // count=1 (valid D#)
  g0.y = lds_off;                                         // lds_addr
  g0.z = (unsigned)(ga & 0xFFFFFFFFu);                    // global_addr[31:0]
  g0.w = (unsigned)((ga >> 32) & 0x1FFFFFFu) | (2u << 30);// addr[56:32], type=2

  i32x8 g1;
  g1[0] = (int)(2u << 16);                    // workgroup_mask=0, data_size=4B
  g1[1] = (int)((total_elems & 0xFFFFu) << 16);  // tensor_dim0[15:0]
  g1[2] = (int)(total_elems >> 16);              // tensor_dim0[31:16], dim1=0
  g1[3] = (int)((total_elems & 0xFFFFu) << 16);  // tile_dim0 = total (1-D)
  g1[4] = 0;                                     // tile_dim1/2 unused
  g1[5] = (int)total_elems;                      // dim0_stride (unused for 1-D)
  g1[6] = 0;
  g1[7] = 0;

  i32x4 z4 = {0, 0, 0, 0};
#if defined(__clang_major__) && __clang_major__ >= 23
  i32x8 z8 = {0, 0, 0, 0, 0, 0, 0, 0};
  __builtin_amdgcn_tensor_load_to_lds(g0, g1, z4, z4, z8, 0);
#else
  __builtin_amdgcn_tensor_load_to_lds(g0, g1, z4, z4, 0);
#endif
  __builtin_amdgcn_s_wait_tensorcnt(0);
#else
  (void)lds_off; (void)src; (void)total_elems;
#endif
}

// ------------------------- WMMA submanifold conv ---------------------------
// One wave32 handles a tile of 16 points. For each of the 9 taps it gathers
// the 16 neighbor feature rows into f32 A-fragments (16x4) and multiplies by
// 4x16 weight slices from LDS with V_WMMA_F32_16X16X4_F32, accumulating
// COUT/16 16x16 f32 C fragments.
//
// A layout (32-bit, 16x4, per ISA 7.12.2): lane = half*16 + m,
//   v0 = A[m][2*half], v1 = A[m][2*half+1]  -> one float2 load per lane.
// B layout (symmetric): lane = half*16 + n,
//   v0 = B[2*half][n], v1 = B[2*half+1][n]  -> two ds_load_b32 per lane.
template <int CIN, int COUT>
__global__ __launch_bounds__(256) void subm_conv_wmma(
    const int* __restrict__ indices,   // N x 3
    const float* __restrict__ feats,   // N x CIN
    const float* __restrict__ w,       // 9 x CIN x COUT
    const int* __restrict__ grid,      // GB*GH*GW
    float* __restrict__ out,           // N x COUT
    int N) {
  constexpr int NK = (CIN + 3) / 4;  // K-slices of 4 (CIN=2 zero-padded)
  constexpr int NT = COUT / 16;      // N-tiles of 16
  constexpr int WTOT = 9 * CIN * COUT;

  __shared__ float w_lds[WTOT];      // 2.3KB / 72KB / 144KB (<320KB WGP LDS)

  // Stage weights: TDM async DMA from wave 0, then cooperative-copy backstop
  // (same data; guards against D# field mis-encoding, LDS OOB writes drop).
  if (threadIdx.x == 0) {
    tdm_load_f32((unsigned)(unsigned long long)(&w_lds[0]), w, WTOT);
  }
  __syncthreads();
  for (int i = threadIdx.x; i < WTOT; i += blockDim.x) w_lds[i] = w[i];
  __syncthreads();

  const int lane = threadIdx.x & 31;
  const int wave = threadIdx.x >> 5;
  const int wavesPerBlock = blockDim.x >> 5;
  const int numTiles = (N + 15) >> 4;
  const int m = lane & 15;   // row (point) / col (channel) within tile
  const int half = lane >> 4;

  for (int tile = blockIdx.x * wavesPerBlock + wave; tile < numTiles;
       tile += gridDim.x * wavesPerBlock) {
    const int p = tile * 16 + m;
    const bool pvalid = p < N;
    int bb = 0, yy = 0, xx = 0;
    if (pvalid) {
      bb = indices[p * 3 + 0];
      yy = indices[p * 3 + 1];
      xx = indices[p * 3 + 2];
    }

    v8f acc[NT];
#pragma unroll
    for (int t = 0; t < NT; ++t)
#pragma unroll
      for (int r = 0; r < 8; ++r) acc[t][r] = 0.0f;

    for (int k = 0; k < 9; ++k) {  // 3x3 taps (kept rolled: code size)
      const int dy = k / 3 - 1;
      const int dx = k % 3 - 1;
      const int ny = yy + dy;
      const int nx = xx + dx;
      const bool inb = pvalid & (ny >= 0) & (ny < GH) & (nx >= 0) & (nx < GW);
      int nb = -1;
      if (inb) nb = grid[(bb * GH + ny) * GW + nx];
      const bool fv = inb & (nb >= 0);
      const float* frow = feats + (size_t)(fv ? nb : 0) * CIN;

      for (int kc = 0; kc < NK; ++kc) {
        const int kbase = kc * 4 + half * 2;  // K row pair handled by this lane
        v2f a;
        if (fv & (kbase < CIN)) {
          a.x = frow[kbase];
          a.y = frow[kbase + 1];
        } else {
          a.x = 0.0f;
          a.y = 0.0f;
        }
#pragma unroll
        for (int t = 0; t < NT; ++t) {
          v2f bf;
          if (kbase < CIN) {  // zero-padded K rows for CIN=2
            const float* wl = w_lds + (k * CIN + kbase) * COUT + t * 16 + m;
            bf.x = wl[0];
            bf.y = wl[COUT];
          } else {
            bf.x = 0.0f;
            bf.y = 0.0f;
          }
          acc[t] = __builtin_amdgcn_wmma_f32_16x16x4_f32(
              false, a, false, bf, (short)0, acc[t], false, false);
        }
      }
    }

    // C layout: VGPR r holds M = r + 8*half, N = t*16 + m.
#pragma unroll
    for (int t = 0; t < NT; ++t)
#pragma unroll
      for (int r = 0; r < 8; ++r) {
        const int row = tile * 16 + r + half * 8;
        if (row < N) out[(size_t)row * COUT + t * 16 + m] = acc[t][r];
      }
  }
}

// ------------------------------ batch norm ---------------------------------
__global__ void bn_zero_stats(float* __restrict__ stats) {
  if (threadIdx.x < 128) stats[threadIdx.x] = 0.0f;  // sum[64] + sumsq[64]
}

template <int C>
__global__ __launch_bounds__(256) void bn_stats(const float* __restrict__ x, int N,
                                                float* __restrict__ sum,
                                                float* __restrict__ sumsq) {
  constexpr int RPB = 256 / C;  // rows per block-pass
  const int c = threadIdx.x % C;
  const int rl = threadIdx.x / C;
  float s = 0.0f, s2 = 0.0f;
  for (long long row = (long long)blockIdx.x * RPB + rl; row < N;
       row += (long long)gridDim.x * RPB) {
    float v = x[row * C + c];
    s += v;
    s2 += v * v;
  }
  __shared__ float ls[256];
  __shared__ float ls2[256];
  ls[threadIdx.x] = s;
  ls2[threadIdx.x] = s2;
  __syncthreads();
  if (threadIdx.x < C) {
    float ts = 0.0f, ts2 = 0.0f;
#pragma unroll
    for (int i = 0; i < RPB; ++i) {
      ts += ls[i * C + threadIdx.x];
      ts2 += ls2[i * C + threadIdx.x];
    }
    atomicAdd(&sum[threadIdx.x], ts);
    atomicAdd(&sumsq[threadIdx.x], ts2);
  }
}

__global__ void bn_finalize(const float* __restrict__ sum,
                            const float* __restrict__ sumsq,
                            const float* __restrict__ g, const float* __restrict__ b,
                            float* __restrict__ scale, float* __restrict__ shift,
                            int C, int N) {
  int c = threadIdx.x;
  if (c < C) {
    float mu = sum[c] / (float)N;
    float var = sumsq[c] / (float)N - mu * mu;  // population var (jnp.var)
    float sc = g[c] * rsqrtf(var + EPS);
    scale[c] = sc;
    shift[c] = b[c] - mu * sc;
  }
}

template <int C>
__global__ __launch_bounds__(256) void bn_apply(float* __restrict__ x, int N,
                                                const float* __restrict__ scale,
                                                const float* __restrict__ shift) {
  const long long total = (long long)N * C;
  for (long long i = (long long)blockIdx.x * blockDim.x + threadIdx.x; i < total;
       i += (long long)gridDim.x * blockDim.x) {
    int c = (int)(i % C);
    float y = x[i] * scale[c] + shift[c];
    x[i] = y >= 0.0f ? y : SLOPE * y;
  }
}

// ------------------------------- misc copy ---------------------------------
__global__ __launch_bounds__(256) void copy_i32(const int* __restrict__ src,
                                                int* __restrict__ dst, int n) {
  for (int i = blockIdx.x * blockDim.x + threadIdx.x; i < n;
       i += gridDim.x * blockDim.x)
    dst[i] = src[i];
}

// ------------------------------- launcher ----------------------------------
extern "C" void kernel_launch(void* const* d_in, const int* in_sizes, int n_in,
                              void* d_out, int out_size, void* d_ws, size_t ws_size,
                              hipStream_t stream) {
  const int* indices = (const int*)d_in[0];
  const float* features = (const float*)d_in[1];
  const float* w1 = (const float*)d_in[2];
  const float* w2 = (const float*)d_in[3];
  const float* w3 = (const float*)d_in[4];
  const float* g1 = (const float*)d_in[5];
  const float* b1 = (const float*)d_in[6];
  const float* g2 = (const float*)d_in[7];
  const float* b2 = (const float*)d_in[8];
  const float* g3 = (const float*)d_in[9];
  const float* b3 = (const float*)d_in[10];

  const int N = in_sizes[0] / 3;
  const int gridTotal = GB * GH * GW;

  // workspace layout
  char* ws = (char*)d_ws;
  int* grid = (int*)ws;                                   // 8 MB
  float* x1 = (float*)(ws + (size_t)gridTotal * 4);       // N*32 f32
  float* x2 = x1 + (size_t)N * 32;                        // N*64 f32
  float* stats = x2 + (size_t)N * 64;                     // sum/sumsq/scale/shift
  float* ssum = stats;
  float* ssq = stats + 64;
  float* scl = stats + 128;
  float* shf = stats + 192;

  int* out_idx = (int*)d_out;
  float* x3 = (float*)d_out + (size_t)N * 3;  // final features live in d_out

  // 1) build hash grid
  grid_init<<<2048, 256, 0, stream>>>(grid, gridTotal);
  grid_scatter<<<(N + 255) / 256, 256, 0, stream>>>(indices, grid, N);

  const int convBlocks = 640;   // 640 blocks * 8 waves ~ 5120 tiles in flight
  const int statBlocks = 1024;
  const int ewBlocks = 2048;

  // 2) layer 1: conv(2->32) + BN + LeakyReLU
  subm_conv_wmma<2, 32><<<convBlocks, 256, 0, stream>>>(indices, features, w1,
                                                        grid, x1, N);
  bn_zero_stats<<<1, 128, 0, stream>>>(stats);
  bn_stats<32><<<statBlocks, 256, 0, stream>>>(x1, N, ssum, ssq);
  bn_finalize<<<1, 64, 0, stream>>>(ssum, ssq, g1, b1, scl, shf, 32, N);
  bn_apply<32><<<ewBlocks, 256, 0, stream>>>(x1, N, scl, shf);

  // 3) layer 2: conv(32->64) + BN + LeakyReLU
  subm_conv_wmma<32, 64><<<convBlocks, 256, 0, stream>>>(indices, x1, w2, grid,
                                                         x2, N);
  bn_zero_stats<<<1, 128, 0, stream>>>(stats);
  bn_stats<64><<<statBlocks, 256, 0, stream>>>(x2, N, ssum, ssq);
  bn_finalize<<<1, 64, 0, stream>>>(ssum, ssq, g2, b2, scl, shf, 64, N);
  bn_apply<64><<<ewBlocks, 256, 0, stream>>>(x2, N, scl, shf);

  // 4) layer 3: conv(64->64) straight into d_out, BN in place
  subm_conv_wmma<64, 64><<<convBlocks, 256, 0, stream>>>(indices, x2, w3, grid,
                                                         x3, N);
  bn_zero_stats<<<1, 128, 0, stream>>>(stats);
  bn_stats<64><<<statBlocks, 256, 0, stream>>>(x3, N, ssum, ssq);
  bn_finalize<<<1, 64, 0, stream>>>(ssum, ssq, g3, b3, scl, shf, 64, N);
  bn_apply<64><<<ewBlocks, 256, 0, stream>>>(x3, N, scl, shf);

  // 5) indices pass-through (bit-copy int32 into the float-typed out buffer)
  copy_i32<<<(3 * N + 255) / 256, 256, 0, stream>>>(indices, out_idx, 3 * N);
}